// LSTMModel_33088428048997
// MI455X (gfx1250) — compile-verified
//
#include <hip/hip_runtime.h>

typedef __attribute__((ext_vector_type(2))) float v2f;
typedef __attribute__((ext_vector_type(8))) float v8f;

constexpr int Bn = 64;        // batch
constexpr int Sn = 512;       // sequence length
constexpr int Dn = 256;       // input dim
constexpr int Hn = 1024;      // hidden dim
constexpr int G4 = 4 * Hn;    // 4096 gate width
constexpr int On = 256;       // fc output dim
constexpr int HROW = 1028;    // padded LDS row stride (floats): 16 A-frag lanes
                              // land on banks 4 apart -> conflict-free
constexpr size_t SCAN_LDS_BYTES = (size_t)Bn * HROW * 4;   // 257 KB (<320 KB/WG)

static __device__ __forceinline__ v8f wmma_f32(v2f a, v2f b, v8f c) {
    // V_WMMA_F32_16X16X4_F32 : D = A(16x4) x B(4x16) + C(16x16), fp32
    return __builtin_amdgcn_wmma_f32_16x16x4_f32(
        false, a, false, b, (short)0, c, false, false);
}

static __device__ __forceinline__ float fsigmoid(float x) {
    return 1.0f / (1.0f + __expf(-x));
}
static __device__ __forceinline__ float ftanh(float x) {
    return 1.0f - 2.0f / (__expf(2.0f * x) + 1.0f);
}

// ---------------------------------------------------------------------------
// Gate pre-projection GEMM, output TRANSPOSED per step:
//   Gt[s][n][b] = X_row(s,b) . W[n, :] + b1[n] + b2[n]      (Gt is (S, 4H, B))
// Source row of X for (s,b) is X + s*xs_s + b*xs_b (handles x=(B,S,D) and
// h1=(S,B,H)). Block = 256 threads = 8 waves; wave computes a 32x32 tile
// (2x2 WMMA, fp32 16x16x4) with a double-buffered k pipeline.
// ---------------------------------------------------------------------------
__global__ void __launch_bounds__(256) gate_gemm(
    const float* __restrict__ X, const float* __restrict__ W,
    const float* __restrict__ b1, const float* __restrict__ b2,
    float* __restrict__ Gt, int K, int xs_s, int xs_b)
{
    const int lane  = threadIdx.x & 31;
    const int wave  = threadIdx.x >> 5;
    const int waveM = wave & 3;   // 4 waves along M
    const int waveN = wave >> 2;  // 2 waves along N
    const int m_base = blockIdx.y * 128 + waveM * 32;
    const int n_base = blockIdx.x * 64  + waveN * 32;

    const int lrow  = lane & 15;
    const int khalf = (lane >> 4) * 2;   // which K-pair this lane holds

    // A fragment row pointers (two 16-row tiles). m = s*64 + b.
    const int m0 = m_base + lrow;
    const int m1 = m0 + 16;
    const float* a0p = X + (long)(m0 >> 6) * xs_s + (long)(m0 & 63) * xs_b + khalf;
    const float* a1p = X + (long)(m1 >> 6) * xs_s + (long)(m1 & 63) * xs_b + khalf;

    // B fragment col pointers (two 16-col tiles); B[k][n] = W[n][k].
    const int n0 = n_base + lrow;
    const int n1 = n0 + 16;
    const float* b0p = W + (long)n0 * K + khalf;
    const float* b1p = W + (long)n1 * K + khalf;

    v8f c00 = {}; v8f c01 = {}; v8f c10 = {}; v8f c11 = {};

    // Double-buffered pipeline: load k+4 while WMMA-ing k.
    v2f a0[2], a1[2], bb0[2], bb1[2];
    auto load_frag = [&](int buf, int k) {
        a0[buf]  = *(const v2f*)(a0p + k);
        a1[buf]  = *(const v2f*)(a1p + k);
        bb0[buf] = *(const v2f*)(b0p + k);
        bb1[buf] = *(const v2f*)(b1p + k);
    };
    auto do_wmma = [&](int buf) {
        c00 = wmma_f32(a0[buf], bb0[buf], c00);
        c01 = wmma_f32(a0[buf], bb1[buf], c01);
        c10 = wmma_f32(a1[buf], bb0[buf], c10);
        c11 = wmma_f32(a1[buf], bb1[buf], c11);
    };

    load_frag(0, 0);
    for (int k = 0; k < K - 8; k += 8) {
        load_frag(1, k + 4);
        do_wmma(0);
        load_frag(0, k + 8);
        do_wmma(1);
    }
    load_frag(1, K - 4);
    do_wmma(0);
    do_wmma(1);

    const float biasA = b1[n0] + b2[n0];
    const float biasB = b1[n1] + b2[n1];

    // Vectorized transposed stores: 8 C-rows per lane are contiguous in b.
    const int rbase = (lane >> 4) * 8;
    const int mrow  = m_base + rbase;       // first row of this lane's block
    const int sA    = mrow >> 6;            // same s for all 8 rows (+16 tile)
    const int bA    = mrow & 63;
    float* g0 = Gt + ((long)sA * G4 + n0) * Bn + bA;
    float* g1 = Gt + ((long)sA * G4 + n1) * Bn + bA;
    *(v8f*)(g0)      = c00 + biasA;
    *(v8f*)(g0 + 16) = c10 + biasA;
    *(v8f*)(g1)      = c01 + biasB;
    *(v8f*)(g1 + 16) = c11 + biasB;
}

// ---------------------------------------------------------------------------
// Grid-wide barrier (monotonic counter; counter zeroed before each scan).
// ---------------------------------------------------------------------------
static __device__ __forceinline__ void grid_sync(unsigned* cnt, unsigned expected) {
    __threadfence();      // release h stores to device scope
    __syncthreads();
    if (threadIdx.x == 0) {
        __hip_atomic_fetch_add(cnt, 1u, __ATOMIC_RELEASE, __HIP_MEMORY_SCOPE_AGENT);
        while (__hip_atomic_load(cnt, __ATOMIC_ACQUIRE, __HIP_MEMORY_SCOPE_AGENT) < expected)
            __builtin_amdgcn_s_sleep(1);
    }
    __syncthreads();
    __threadfence();      // acquire: don't read stale h from near caches
}

// ---------------------------------------------------------------------------
// Persistent LSTM scan. Grid = 8 blocks x 256 threads = 64 waves.
// Wave gw owns hidden columns [gw*16, gw*16+16). Cell state c stays in VGPRs
// for all 512 steps. Per step:
//   1) block cooperatively stages h_{s-1} (64x1024, 256 KB) into LDS
//      (one copy per block instead of one per wave: 8x less L2 traffic),
//   2) fp32 WMMA GEMM: A-fragments from LDS (ds_load_b64, conflict-free via
//      padded row stride), B-fragments (w_hh) streamed from L2,
//   3) gate nonlinearity + cell update in registers, h stored to global,
//   4) grid-wide barrier.
// ---------------------------------------------------------------------------
__global__ void __launch_bounds__(256, 1) lstm_scan(
    const float* __restrict__ Gt,    // (S, 4H, B) pre-projected gates + biases
    const float* __restrict__ Whh,   // (4H, H) row-major
    float* __restrict__ Hout,        // (hmod, B, H)
    unsigned* __restrict__ barrier_cnt,
    int hmod)
{
    extern __shared__ float hbuf[];            // Bn rows x HROW floats (padded)

    const int tid   = threadIdx.x;
    const int lane  = tid & 31;
    const int wave  = tid >> 5;
    const int gw    = blockIdx.x * 8 + wave;   // 0..63
    const int lrow  = lane & 15;
    const int khalf = (lane >> 4) * 2;
    const int rbase = (lane >> 4) * 8;
    const int ncol  = gw * 16 + lrow;          // hidden column for this lane

    // B fragment pointers: one per gate, fixed for the whole scan.
    const float* wptr[4];
#pragma unroll
    for (int g = 0; g < 4; ++g)
        wptr[g] = Whh + (long)(g * Hn + ncol) * Hn + khalf;

    // A fragment pointers into LDS, fixed for the whole scan.
    const float* aptr[4];
#pragma unroll
    for (int mt = 0; mt < 4; ++mt)
        aptr[mt] = hbuf + (mt * 16 + lrow) * HROW + khalf;

    v8f cst[4] = {v8f{}, v8f{}, v8f{}, v8f{}};   // cell state, 4 M-tiles

    for (int s = 0; s < Sn; ++s) {
        // Init accumulators from pre-projected gates: contiguous v8f loads.
        v8f acc[4][4];
        {
            const float* gb = Gt + (long)s * G4 * Bn;
#pragma unroll
            for (int g = 0; g < 4; ++g)
#pragma unroll
                for (int mt = 0; mt < 4; ++mt)
                    acc[g][mt] = *(const v8f*)(gb + ((long)g * Hn + ncol) * Bn
                                                  + mt * 16 + rbase);
        }

        if (s > 0) {
            // Stage h_{s-1} into LDS: 64 rows x 1024 floats, float4-wide.
            const float* hprev = Hout + (long)((s - 1) % hmod) * Bn * Hn;
#pragma unroll 8
            for (int j = 0; j < Bn; ++j) {
                const float4 v = *(const float4*)(hprev + (long)j * Hn + tid * 4);
                *(float4*)(&hbuf[j * HROW + tid * 4]) = v;
            }
            __syncthreads();

            v2f af[2][4], bf[2][4];
            auto load_frag = [&](int buf, int k) {
#pragma unroll
                for (int mt = 0; mt < 4; ++mt) af[buf][mt] = *(const v2f*)(aptr[mt] + k);
#pragma unroll
                for (int g = 0; g < 4; ++g)    bf[buf][g]  = *(const v2f*)(wptr[g] + k);
            };
            auto do_wmma = [&](int buf) {
#pragma unroll
                for (int g = 0; g < 4; ++g)
#pragma unroll
                    for (int mt = 0; mt < 4; ++mt)
                        acc[g][mt] = wmma_f32(af[buf][mt], bf[buf][g], acc[g][mt]);
            };

            load_frag(0, 0);
            for (int k = 0; k < Hn - 8; k += 8) {
                load_frag(1, k + 4);
                do_wmma(0);
                load_frag(0, k + 8);
                do_wmma(1);
            }
            load_frag(1, Hn - 4);
            do_wmma(0);
            do_wmma(1);
        }

        // Gate nonlinearity + state update + h store.
        float* hcur = Hout + (long)(s % hmod) * Bn * Hn;
#pragma unroll
        for (int mt = 0; mt < 4; ++mt) {
#pragma unroll
            for (int r = 0; r < 8; ++r) {
                const float iv = fsigmoid(acc[0][mt][r]);
                const float fv = fsigmoid(acc[1][mt][r]);
                const float gv = ftanh(acc[2][mt][r]);
                const float ov = fsigmoid(acc[3][mt][r]);
                const float cv = fv * cst[mt][r] + iv * gv;
                cst[mt][r] = cv;
                hcur[(long)(mt * 16 + rbase + r) * Hn + ncol] = ov * ftanh(cv);
            }
        }

        if (s + 1 < Sn)
            grid_sync(barrier_cnt, (unsigned)gridDim.x * (unsigned)(s + 1));
    }
}

// ---------------------------------------------------------------------------
// Final FC: out[b,o] = h_last[b,:] . w_fc[o,:] + b_fc[o]  (tiny: 33 MFLOP)
// ---------------------------------------------------------------------------
__global__ void fc_kernel(const float* __restrict__ h,
                          const float* __restrict__ Wfc,
                          const float* __restrict__ bfc,
                          float* __restrict__ out)
{
    const int o = threadIdx.x;   // 0..255
    const int b = blockIdx.x;    // 0..63
    const float* hp = h   + (long)b * Hn;
    const float* wp = Wfc + (long)o * Hn;
    float acc = 0.0f;
    for (int k = 0; k < Hn; k += 4) {
        const float4 hv = *(const float4*)(hp + k);
        const float4 wv = *(const float4*)(wp + k);
        acc += hv.x * wv.x + hv.y * wv.y + hv.z * wv.z + hv.w * wv.w;
    }
    out[(long)b * On + o] = acc + bfc[o];
}

__global__ void zero_kernel(unsigned* p, int n) {
    const int i = blockIdx.x * blockDim.x + threadIdx.x;
    if (i < n) p[i] = 0u;
}

// ---------------------------------------------------------------------------
extern "C" void kernel_launch(void* const* d_in, const int* in_sizes, int n_in,
                              void* d_out, int out_size, void* d_ws, size_t ws_size,
                              hipStream_t stream) {
    (void)in_sizes; (void)n_in; (void)out_size; (void)ws_size;
    const float* x    = (const float*)d_in[0];
    const float* wih0 = (const float*)d_in[1];
    const float* whh0 = (const float*)d_in[2];
    const float* bih0 = (const float*)d_in[3];
    const float* bhh0 = (const float*)d_in[4];
    const float* wih1 = (const float*)d_in[5];
    const float* whh1 = (const float*)d_in[6];
    const float* bih1 = (const float*)d_in[7];
    const float* bhh1 = (const float*)d_in[8];
    const float* wfc  = (const float*)d_in[9];
    const float* bfc  = (const float*)d_in[10];
    float* out = (float*)d_out;

    char* ws = (char*)d_ws;
    unsigned* cnt = (unsigned*)ws;                              // barrier counter
    float* Gbuf = (float*)(ws + 1024);                          // (S,4H,B) 512 MB, reused
    float* H1   = (float*)(ws + 1024 + (size_t)Sn * Bn * G4 * 4);           // (S,B,H) 128 MB
    float* H2   = (float*)(ws + 1024 + (size_t)Sn * Bn * G4 * 4
                                     + (size_t)Sn * Bn * Hn * 4);           // (2,B,H) 512 KB

    const dim3 blk(256);
    const dim3 gemm_grid(G4 / 64, (Sn * Bn) / 128);

    // ---- Layer 0 ----
    zero_kernel<<<1, 64, 0, stream>>>(cnt, 64);
    // x is (B,S,D): row (s,b) at s*D + b*S*D
    gate_gemm<<<gemm_grid, blk, 0, stream>>>(x, wih0, bih0, bhh0, Gbuf,
                                             Dn, /*xs_s=*/Dn, /*xs_b=*/Sn * Dn);
    lstm_scan<<<dim3(8), blk, SCAN_LDS_BYTES, stream>>>(Gbuf, whh0, H1, cnt,
                                                        /*hmod=*/Sn);

    // ---- Layer 1 ----
    zero_kernel<<<1, 64, 0, stream>>>(cnt, 64);
    // H1 is (S,B,H): row (s,b) at s*B*H + b*H
    gate_gemm<<<gemm_grid, blk, 0, stream>>>(H1, wih1, bih1, bhh1, Gbuf,
                                             Hn, /*xs_s=*/Bn * Hn, /*xs_b=*/Hn);
    lstm_scan<<<dim3(8), blk, SCAN_LDS_BYTES, stream>>>(Gbuf, whh1, H2, cnt,
                                                        /*hmod=*/2);

    // ---- FC on last h2 (slot (S-1)%2 == 1) ----
    fc_kernel<<<dim3(Bn), dim3(On), 0, stream>>>(H2 + (size_t)((Sn - 1) % 2) * Bn * Hn,
                                                 wfc, bfc, out);
}